// GAT_15985868276244
// MI455X (gfx1250) — compile-verified
//
#include <hip/hip_runtime.h>
#include <math.h>

typedef __attribute__((ext_vector_type(2))) float v2f;
typedef __attribute__((ext_vector_type(8))) float v8f;

#define NEG_SLOPE 0.2f
#define GAT_EPS 1e-16f

// ---------------------------------------------------------------------------
// helpers: ordered-uint encoding of float for atomicMax (monotone map)
// ---------------------------------------------------------------------------
__device__ __forceinline__ unsigned f2ord(float f) {
  unsigned u = __float_as_uint(f);
  return (u >> 31) ? ~u : (u | 0x80000000u);
}
__device__ __forceinline__ float ord2f(unsigned t) {
  unsigned u = (t >> 31) ? (t & 0x7FFFFFFFu) : ~t;
  return __uint_as_float(u);
}
#define ORD_NEG_INF 0x007FFFFFu  // f2ord(-inf)

// ---------------------------------------------------------------------------
// fill kernels
// ---------------------------------------------------------------------------
__global__ void fill_f32(float* __restrict__ p, float v, long long n) {
  long long stride = (long long)gridDim.x * blockDim.x;
  for (long long i = (long long)blockIdx.x * blockDim.x + threadIdx.x; i < n; i += stride)
    p[i] = v;
}
__global__ void fill_u32(unsigned* __restrict__ p, unsigned v, long long n) {
  long long stride = (long long)gridDim.x * blockDim.x;
  for (long long i = (long long)blockIdx.x * blockDim.x + threadIdx.x; i < n; i += stride)
    p[i] = v;
}

// ---------------------------------------------------------------------------
// C[M, NC] = X[M, K] @ W[NC, K]^T   via V_WMMA_F32_16X16X4_F32, wave per tile
// M must be a multiple of 16 (50000 = 16*3125). K multiple of 4.
// ---------------------------------------------------------------------------
__global__ __launch_bounds__(256) void gemm_wmma_nt(
    const float* __restrict__ X, const float* __restrict__ W,
    float* __restrict__ C, int Mtiles, int Ntiles, int K, int NC) {
  int wave = blockIdx.x * (blockDim.x >> 5) + (threadIdx.x >> 5);
  if (wave >= Mtiles * Ntiles) return;
  int tm = wave / Ntiles;
  int tn = wave % Ntiles;
  int lane = threadIdx.x & 31;
  int hf = lane >> 4;     // 0: K pair {0,1}, 1: K pair {2,3}
  int lm = lane & 15;     // M (for A) / N (for B) within tile

  const float* xrow = X + (size_t)(tm * 16 + lm) * K;
  const float* wrow = W + (size_t)(tn * 16 + lm) * K;

  v8f acc = {0.f, 0.f, 0.f, 0.f, 0.f, 0.f, 0.f, 0.f};
  for (int k = 0; k < K; k += 4) {
    int ka = k + hf * 2;
    v2f a, b;
    a.x = xrow[ka]; a.y = xrow[ka + 1];
    b.x = wrow[ka]; b.y = wrow[ka + 1];
    acc = __builtin_amdgcn_wmma_f32_16x16x4_f32(false, a, false, b,
                                                (short)0, acc, false, false);
  }
  // C/D layout: VGPR i -> M = i + hf*8, N = lm
  int col = tn * 16 + lm;
  int row = tm * 16 + hf * 8;
#pragma unroll
  for (int i = 0; i < 8; ++i)
    C[(size_t)(row + i) * NC + col] = acc[i];
}

// ---------------------------------------------------------------------------
// per-(node, head) attention logits: a_s = <xl[n,h,:], att_src[h,:]>, same a_d
// ---------------------------------------------------------------------------
__global__ void node_att(const float* __restrict__ xl,
                         const float* __restrict__ att_src,
                         const float* __restrict__ att_dst,
                         float* __restrict__ as_, float* __restrict__ ad_,
                         int N, int H, int CH) {
  int t = blockIdx.x * blockDim.x + threadIdx.x;
  if (t >= N * H) return;
  int n = t / H, h = t % H;
  const float* v = xl + (size_t)n * H * CH + (size_t)h * CH;
  const float* s = att_src + h * CH;
  const float* d = att_dst + h * CH;
  float accs = 0.f, accd = 0.f;
#pragma unroll 4
  for (int c = 0; c < CH; ++c) {
    float x = v[c];
    accs += x * s[c];
    accd += x * d[c];
  }
  as_[t] = accs;
  ad_[t] = accd;
}

// ---------------------------------------------------------------------------
// per-(edge, head): alpha = leaky_relu(a_s[src]+a_d[dst]); segment max (atomic)
// edge e < E: src=ei[e], dst=ei[E+e]; e >= E: self loop node (e-E)
// ---------------------------------------------------------------------------
__global__ void edge_alpha(const long long* __restrict__ ei,
                           const float* __restrict__ as_,
                           const float* __restrict__ ad_,
                           float* __restrict__ ealpha,
                           unsigned* __restrict__ amax,
                           int E, int N, int H) {
  long long t = (long long)blockIdx.x * blockDim.x + threadIdx.x;
  long long tot = (long long)(E + N) * H;
  if (t >= tot) return;
  long long e = t / H;
  int h = (int)(t % H);
  int src, dst;
  if (e < E) { src = (int)ei[e]; dst = (int)ei[E + e]; }
  else       { src = dst = (int)(e - E); }
  float a = as_[(size_t)src * H + h] + ad_[(size_t)dst * H + h];
  a = (a > 0.f) ? a : NEG_SLOPE * a;
  ealpha[t] = a;
  atomicMax(&amax[(size_t)dst * H + h], f2ord(a));
}

// ---------------------------------------------------------------------------
// per-(edge, head): ex = exp(alpha - amax[dst]); segment sum (atomic)
// ---------------------------------------------------------------------------
__global__ void edge_expsum(const long long* __restrict__ ei,
                            float* __restrict__ ealpha,
                            const unsigned* __restrict__ amax,
                            float* __restrict__ denom,
                            int E, int N, int H) {
  long long t = (long long)blockIdx.x * blockDim.x + threadIdx.x;
  long long tot = (long long)(E + N) * H;
  if (t >= tot) return;
  long long e = t / H;
  int h = (int)(t % H);
  int dst;
  if (e < E) dst = (int)ei[E + e];
  else       dst = (int)(e - E);
  float m = ord2f(amax[(size_t)dst * H + h]);
  float ex = expf(ealpha[t] - m);
  ealpha[t] = ex;
  atomicAdd(&denom[(size_t)dst * H + h], ex);
}

// ---------------------------------------------------------------------------
// wave per edge: out[dst, :] += xl[src, :] * coef[head]. PER = channels/lane.
// C = H*CH total channels; PER*32 == C; CH multiple of PER*?? (4 | 64 holds)
// ---------------------------------------------------------------------------
template <int PER>
__global__ __launch_bounds__(256) void edge_scatter(
    const long long* __restrict__ ei, const float* __restrict__ xl,
    const float* __restrict__ ealpha, const float* __restrict__ denom,
    float* __restrict__ out, int E, int N, int H, int CH) {
  long long wid = (long long)blockIdx.x * (blockDim.x >> 5) + (threadIdx.x >> 5);
  long long tot = (long long)E + N;
  if (wid >= tot) return;
  int lane = threadIdx.x & 31;
  int src, dst;
  if (wid < E) { src = (int)ei[wid]; dst = (int)ei[E + wid]; }
  else         { src = dst = (int)(wid - E); }
  int C = H * CH;
  int c0 = lane * PER;
  int h = c0 / CH;
  float ex = ealpha[wid * H + h];
  float den = denom[(size_t)dst * H + h];
  float coef = ex / (den + GAT_EPS);
  const float* xs = xl + (size_t)src * C + c0;
  float* od = out + (size_t)dst * C + c0;
  float vals[PER];
#pragma unroll
  for (int i = 0; i < PER; ++i) vals[i] = xs[i] * coef;
#pragma unroll
  for (int i = 0; i < PER; ++i) atomicAdd(&od[i], vals[i]);
}

// ---------------------------------------------------------------------------
// h = elu(h + b)   (per element)
// ---------------------------------------------------------------------------
__global__ void bias_elu(float* __restrict__ hbuf, const float* __restrict__ b,
                         long long n, int C) {
  long long t = (long long)blockIdx.x * blockDim.x + threadIdx.x;
  if (t >= n) return;
  int c = (int)(t % C);
  float v = hbuf[t] + b[c];
  hbuf[t] = (v > 0.f) ? v : (expf(v) - 1.f);
}

// ---------------------------------------------------------------------------
// wave per node: row softmax over C=64 channels (2 per lane), with bias add
// ---------------------------------------------------------------------------
__global__ __launch_bounds__(256) void bias_softmax64(
    float* __restrict__ out, const float* __restrict__ b, int N) {
  int wid = blockIdx.x * (blockDim.x >> 5) + (threadIdx.x >> 5);
  if (wid >= N) return;
  int lane = threadIdx.x & 31;
  float* row = out + (size_t)wid * 64;
  float v0 = row[lane] + b[lane];
  float v1 = row[lane + 32] + b[lane + 32];
  float m = fmaxf(v0, v1);
#pragma unroll
  for (int o = 16; o > 0; o >>= 1) m = fmaxf(m, __shfl_xor(m, o, 32));
  float e0 = expf(v0 - m), e1 = expf(v1 - m);
  float s = e0 + e1;
#pragma unroll
  for (int o = 16; o > 0; o >>= 1) s += __shfl_xor(s, o, 32);
  float inv = 1.f / s;
  row[lane] = e0 * inv;
  row[lane + 32] = e1 * inv;
}

// ---------------------------------------------------------------------------
// launcher
// ---------------------------------------------------------------------------
extern "C" void kernel_launch(void* const* d_in, const int* in_sizes, int n_in,
                              void* d_out, int out_size, void* d_ws, size_t ws_size,
                              hipStream_t stream) {
  const int F_IN = 128, HID = 64, H1 = 2, OUT = 64;
  const int C1 = H1 * HID;               // 128
  const int N = in_sizes[0] / F_IN;      // 50000
  const int E = in_sizes[1] / 2;         // 800000
  const long long ET = (long long)E + N; // with self-loops

  const float* x  = (const float*)d_in[0];
  const long long* ei = (const long long*)d_in[1];
  const float* W1 = (const float*)d_in[2];
  const float* att_s1 = (const float*)d_in[3];
  const float* att_d1 = (const float*)d_in[4];
  const float* b1 = (const float*)d_in[5];
  const float* W2 = (const float*)d_in[6];
  const float* att_s2 = (const float*)d_in[7];
  const float* att_d2 = (const float*)d_in[8];
  const float* b2 = (const float*)d_in[9];
  float* out = (float*)d_out;

  // workspace carve-up (floats)
  float* ws = (float*)d_ws;
  size_t off = 0;
  auto alloc = [&](size_t n) { float* p = ws + off; off += n; return p; };
  float* xl1  = alloc((size_t)N * C1);
  float* hbuf = alloc((size_t)N * C1);
  float* a1s  = alloc((size_t)N * H1);
  float* a1d  = alloc((size_t)N * H1);
  unsigned* amax1 = (unsigned*)alloc((size_t)N * H1);
  float* den1 = alloc((size_t)N * H1);
  float* ea1  = alloc((size_t)ET * H1);
  float* xl2  = alloc((size_t)N * OUT);
  float* a2s  = alloc((size_t)N);
  float* a2d  = alloc((size_t)N);
  unsigned* amax2 = (unsigned*)alloc((size_t)N);
  float* den2 = alloc((size_t)N);
  float* ea2  = alloc((size_t)ET);
  (void)ws_size;

  auto cdiv = [](long long a, long long b) { return (int)((a + b - 1) / b); };
  const int TB = 256;

  // init accumulators
  fill_f32<<<2048, TB, 0, stream>>>(hbuf, 0.f, (long long)N * C1);
  fill_f32<<<2048, TB, 0, stream>>>(out, 0.f, (long long)N * OUT);
  fill_f32<<<256, TB, 0, stream>>>(den1, 0.f, (long long)N * H1);
  fill_f32<<<256, TB, 0, stream>>>(den2, 0.f, (long long)N);
  fill_u32<<<256, TB, 0, stream>>>(amax1, ORD_NEG_INF, (long long)N * H1);
  fill_u32<<<256, TB, 0, stream>>>(amax2, ORD_NEG_INF, (long long)N);

  // ---------------- layer 1 ----------------
  {
    int Mt = N / 16, Nt = C1 / 16;                    // 3125 x 8
    gemm_wmma_nt<<<cdiv((long long)Mt * Nt, 8), TB, 0, stream>>>(
        x, W1, xl1, Mt, Nt, F_IN, C1);
  }
  node_att<<<cdiv((long long)N * H1, TB), TB, 0, stream>>>(
      xl1, att_s1, att_d1, a1s, a1d, N, H1, HID);
  edge_alpha<<<cdiv(ET * H1, TB), TB, 0, stream>>>(ei, a1s, a1d, ea1, amax1, E, N, H1);
  edge_expsum<<<cdiv(ET * H1, TB), TB, 0, stream>>>(ei, ea1, amax1, den1, E, N, H1);
  edge_scatter<4><<<cdiv(ET * 32, TB), TB, 0, stream>>>(
      ei, xl1, ea1, den1, hbuf, E, N, H1, HID);
  bias_elu<<<cdiv((long long)N * C1, TB), TB, 0, stream>>>(hbuf, b1, (long long)N * C1, C1);

  // ---------------- layer 2 ----------------
  {
    int Mt = N / 16, Nt = OUT / 16;                   // 3125 x 4
    gemm_wmma_nt<<<cdiv((long long)Mt * Nt, 8), TB, 0, stream>>>(
        hbuf, W2, xl2, Mt, Nt, C1, OUT);
  }
  node_att<<<cdiv((long long)N, TB), TB, 0, stream>>>(
      xl2, att_s2, att_d2, a2s, a2d, N, 1, OUT);
  edge_alpha<<<cdiv(ET, TB), TB, 0, stream>>>(ei, a2s, a2d, ea2, amax2, E, N, 1);
  edge_expsum<<<cdiv(ET, TB), TB, 0, stream>>>(ei, ea2, amax2, den2, E, N, 1);
  edge_scatter<2><<<cdiv(ET * 32, TB), TB, 0, stream>>>(
      ei, xl2, ea2, den2, out, E, N, 1, OUT);
  bias_softmax64<<<cdiv((long long)N * 32, TB), TB, 0, stream>>>(out, b2, N);
}